// ConvGen092_65481071405175
// MI455X (gfx1250) — compile-verified
//
#include <hip/hip_runtime.h>

typedef float v2f __attribute__((ext_vector_type(2)));
typedef float v8f __attribute__((ext_vector_type(8)));

#define BN_EPS    1e-5f
#define P1_BLOCKS 510     // 510 blk * 4 waves * 8 rows = 16320 = 64*255 feat rows
#define R3        6       // output rows per wave in fused pass
#define G3        43      // ceil(254 / R3) row groups

// tanh(f) for f >= 0 via exact identity tanh = 1 - 2/(exp(2f)+1).
// v_exp_f32 + v_rcp_f32: ~5 branch-free VALU ops, no EXEC divergence.
__device__ __forceinline__ float tanh_nonneg(float f) {
    float e = __expf(f + f);
    return 1.0f - 2.0f * __builtin_amdgcn_rcpf(e + 1.0f);
}

// ---------------------------------------------------------------------------
// Pass 1: conv stats on the matrix pipe. Each wave: 8 feat rows x 16 col-tiles
// = 128 WMMA tiles (3 chained K=4 WMMAs each), accumulating per-lane
// sum / sumsq of the (bias-free) conv output in registers. One butterfly
// reduction at the end, then fixed-order block + global reduction.
// part[blk*32 + c] = sum_c, part[blk*32 + 16 + c] = sumsq_c  (partial)
// ---------------------------------------------------------------------------
__global__ __launch_bounds__(128) void cg_stats_wmma(
    const float* __restrict__ x, const float* __restrict__ w,
    float* __restrict__ part)
{
    __shared__ float red[4][32];
    const int wave = threadIdx.x >> 5;
    const int lane = threadIdx.x & 31;
    const int n    = lane & 15;          // M row (out-ch for A) / pixel (B,D)
    const int half = lane >> 4;          // K half
    const int gw   = blockIdx.x * 4 + wave;   // 0..2039

    // A operand: lane m=n holds w[m][ci][half][0..1] (K = half*2, half*2+1)
    v2f A[3];
#pragma unroll
    for (int ci = 0; ci < 3; ++ci)
        A[ci] = *(const v2f*)(w + ((n * 3 + ci) * 2 + half) * 2);

    float sum[8], sq[8];
#pragma unroll
    for (int v = 0; v < 8; ++v) { sum[v] = 0.f; sq[v] = 0.f; }

    for (int rr = gw * 8; rr < gw * 8 + 8; ++rr) {   // linear feat-row index
        int bb = rr / 255;
        int i  = rr - bb * 255;
        const float* xb0 = x + ((size_t)(bb * 3 + 0) * 256 + i + half) * 256;
        const float* xb1 = xb0 + 65536;
        const float* xb2 = xb1 + 65536;
        for (int t16 = 0; t16 < 16; ++t16) {
            int col  = t16 * 16 + n;
            int colc = col > 254 ? 254 : col;        // clamp loads only
            v8f acc = {};
            v2f B;
            B.x = xb0[colc]; B.y = xb0[colc + 1];
            acc = __builtin_amdgcn_wmma_f32_16x16x4_f32(false, A[0], false, B, (short)0, acc, false, false);
            B.x = xb1[colc]; B.y = xb1[colc + 1];
            acc = __builtin_amdgcn_wmma_f32_16x16x4_f32(false, A[1], false, B, (short)0, acc, false, false);
            B.x = xb2[colc]; B.y = xb2[colc + 1];
            acc = __builtin_amdgcn_wmma_f32_16x16x4_f32(false, A[2], false, B, (short)0, acc, false, false);
            if (t16 < 15) {                          // wave-uniform branch
#pragma unroll
                for (int v = 0; v < 8; ++v) {
                    sum[v] += acc[v];
                    sq[v]   = fmaf(acc[v], acc[v], sq[v]);
                }
            } else {                                 // edge tile: mask col 255
                bool valid = col < 255;
#pragma unroll
                for (int v = 0; v < 8; ++v) {
                    float m = valid ? acc[v] : 0.f;
                    sum[v] += m;
                    sq[v]   = fmaf(m, m, sq[v]);
                }
            }
        }
    }

    // butterfly within each 16-lane half -> lanes 0/16 hold channel totals
#pragma unroll
    for (int off = 1; off < 16; off <<= 1) {
#pragma unroll
        for (int v = 0; v < 8; ++v) {
            sum[v] += __shfl_xor(sum[v], off, 32);
            sq[v]  += __shfl_xor(sq[v],  off, 32);
        }
    }
    if (n == 0) {                 // lane 0: ch 0-7, lane 16: ch 8-15
#pragma unroll
        for (int v = 0; v < 8; ++v) {
            red[wave][half * 8 + v]      = sum[v];
            red[wave][16 + half * 8 + v] = sq[v];
        }
    }
    __syncthreads();
    if (threadIdx.x < 32) {       // fixed-order per-block reduction
        float a = red[0][threadIdx.x] + red[1][threadIdx.x] +
                  red[2][threadIdx.x] + red[3][threadIdx.x];
        part[blockIdx.x * 32 + threadIdx.x] = a;
    }
}

// ---------------------------------------------------------------------------
// Pass 1b: BN affine params. Bias cancels under mean subtraction:
// bn(conv+b) = conv*scale + (beta - mean_conv*scale).
// stats[c] = scale, stats[16+c] = shift
// ---------------------------------------------------------------------------
__global__ void cg_stats_finalize(
    const float* __restrict__ part, const float* __restrict__ gamma,
    const float* __restrict__ beta, float* __restrict__ stats)
{
    int c = threadIdx.x;
    if (c >= 16) return;
    float s = 0.f, q = 0.f;
    for (int blk = 0; blk < P1_BLOCKS; ++blk) {   // fixed order: deterministic
        s += part[blk * 32 + c];
        q += part[blk * 32 + 16 + c];
    }
    const float N = 64.0f * 255.0f * 255.0f;
    float mean = s / N;
    float var  = q / N - mean * mean;
    float scl  = gamma[c] * rsqrtf(var + BN_EPS);
    stats[c]      = scl;
    stats[16 + c] = beta[c] - mean * scl;
}

// ---------------------------------------------------------------------------
// Pass 2: fused conv(WMMA) + BN + ReLU + fast tanh + channel sum (shfl_xor 16)
// + 2x2 window (shfl_down 1). Each wave: R3=6 output rows x 15 columns
// (7 feat-row tiles x 3 WMMAs). All branches before WMMAs are wave-uniform.
// ---------------------------------------------------------------------------
__global__ __launch_bounds__(128) void cg_fused_wmma(
    const float* __restrict__ x, const float* __restrict__ w,
    const float* __restrict__ stats, float* __restrict__ out, int nTasks)
{
    int wid = blockIdx.x * (blockDim.x >> 5) + (threadIdx.x >> 5);
    if (wid >= nTasks) return;               // wave-uniform guard
    const int lane = threadIdx.x & 31;
    const int n    = lane & 15;
    const int half = lane >> 4;

    int t   = wid % 17;                      // 17 col tiles of 15 outputs
    int tmp = wid / 17;
    int g   = tmp % G3;                      // row group
    int bb  = tmp / G3;                      // batch
    int i0  = g * R3;
    int j0  = t * 15;
    int col = j0 + n;
    if (col > 254) col = 254;                // clamp loads; stores guarded

    v2f A[3];
#pragma unroll
    for (int ci = 0; ci < 3; ++ci)
        A[ci] = *(const v2f*)(w + ((n * 3 + ci) * 2 + half) * 2);

    float sc[8], sh[8];
#pragma unroll
    for (int v = 0; v < 8; ++v) {
        sc[v] = stats[half * 8 + v];
        sh[v] = stats[16 + half * 8 + v];
    }

    float S[R3 + 1];
#pragma unroll
    for (int rt = 0; rt <= R3; ++rt) {       // feat rows i0 .. i0+R3
        int fr = i0 + rt;
        if (fr > 254) fr = 254;              // clamp (boundary group)
        v8f acc = {};
#pragma unroll
        for (int ci = 0; ci < 3; ++ci) {
            const float* xr =
                x + ((size_t)(bb * 3 + ci) * 256 + fr + half) * 256 + col;
            v2f B;
            B.x = xr[0];
            B.y = xr[1];
            acc = __builtin_amdgcn_wmma_f32_16x16x4_f32(
                false, A[ci], false, B, (short)0, acc, false, false);
        }
        float s = 0.f;
#pragma unroll
        for (int v = 0; v < 8; ++v) {
            float f = fmaf(acc[v], sc[v], sh[v]);   // BN affine
            f = fmaxf(f, 0.f);                      // ReLU
            s += tanh_nonneg(f);                    // branch-free tanh
        }
        S[rt] = s + __shfl_xor(s, 16, 32);          // full 16-channel sum
    }

#pragma unroll
    for (int r = 0; r < R3; ++r) {
        int row = i0 + r;
        if (row >= 254) break;                      // wave-uniform
        float T  = S[r] + S[r + 1];
        float Tn = __shfl_down(T, 1, 32);
        int j = j0 + n;
        if (half == 0 && n < 15 && j < 254)
            out[((size_t)bb * 254 + row) * 254 + j] = (T + Tn) * (1.0f / 64.0f);
    }
}

// ---------------------------------------------------------------------------
extern "C" void kernel_launch(void* const* d_in, const int* in_sizes, int n_in,
                              void* d_out, int out_size, void* d_ws, size_t ws_size,
                              hipStream_t stream)
{
    (void)in_sizes; (void)n_in; (void)out_size; (void)ws_size;
    const float* x     = (const float*)d_in[0];
    const float* w     = (const float*)d_in[1];
    // d_in[2] (conv bias) cancels analytically under BN mean subtraction
    const float* gamma = (const float*)d_in[3];
    const float* beta  = (const float*)d_in[4];
    float* out   = (float*)d_out;
    float* part  = (float*)d_ws;                 // P1_BLOCKS*32 floats
    float* stats = part + P1_BLOCKS * 32;        // 32 floats (scale | shift)

    cg_stats_wmma<<<P1_BLOCKS, 128, 0, stream>>>(x, w, part);
    cg_stats_finalize<<<1, 32, 0, stream>>>(part, gamma, beta, stats);

    const int nTasks = 64 * G3 * 17;             // 46784 wave tasks
    const int blocks = nTasks / 4;               // 4 waves per 128-thr block
    cg_fused_wmma<<<blocks, 128, 0, stream>>>(x, w, stats, out, nTasks);
}